// HyperbolicTemporalAttention_1297080123957
// MI455X (gfx1250) — compile-verified
//
#include <hip/hip_runtime.h>
#include <math.h>

typedef __attribute__((ext_vector_type(2))) float v2f;
typedef __attribute__((ext_vector_type(8))) float v8f;

#define B_   16
#define T_   243
#define J_   24
#define D_   256
#define H_   8
#define DH_  32
#define WIN_ 4
#define KW_  9
#define BJ_  (B_ * J_)          // 384
#define M_TOTAL (BJ_ * T_)      // 93312 rows
#define QKV_ELEMS ((size_t)BJ_ * H_ * T_ * DH_)   // 23,887,872 floats per tensor

#define MB_  64                 // M rows per block (4 WMMA M-tiles)
#define KP_  128                // K columns staged per LDS phase
#define LSTR (KP_ + 2)          // padded LDS row stride (floats)

// ---------------------------------------------------------------------------
// Kernel 1: QKV projection GEMM via V_WMMA_F32_16X16X4_F32.
//   qkv[m, n] = sum_k x[m, k] * W_qkv[n, k] + b_qkv[n]
// Block: 512 threads (16 waves). Block owns 64 rows (4 M-tiles) staged in LDS
// in two K=128 phases; each wave owns 3 N-tiles (16 waves x 3 = all 48).
// B fragments are reused across the 4 M-tiles -> 12 WMMA per 7 loads, and
// W_qkv L2 traffic drops 4x vs an M=16 blocking (1458 blocks x 786KB = 1.15GB).
// Results scattered into ws as q/k/v each laid out [bj][h][t][dh].
// ---------------------------------------------------------------------------
__global__ __launch_bounds__(512)
void qkv_gemm_wmma(const float* __restrict__ x,
                   const float* __restrict__ Wq,
                   const float* __restrict__ bq,
                   float* __restrict__ ws)
{
    __shared__ float sA[MB_][LSTR];   // 64 x 130 x 4B = 33.3 KB

    const int m0   = blockIdx.x * MB_;
    const int tid  = threadIdx.x;
    const int wave = tid >> 5;          // 0..15
    const int lane = tid & 31;
    const int lrow = lane & 15;         // M (for A) / N (for B) within tile
    const int khalf = (lane >> 4) * 2;  // lanes 16..31 hold K+2, K+3

    // Source row pointer for this thread's cooperative-load row.
    const int crow = tid >> 3;          // 0..63 (8 threads per row)
    const int cseg = tid & 7;
    const float* srcRow;
    {
        const int m  = m0 + crow;
        const int bj = m / T_;
        const int t  = m - bj * T_;
        const int b  = bj / J_;
        const int j  = bj - b * J_;
        srcRow = x + (((size_t)b * T_ + t) * J_ + j) * D_;
    }

    v8f acc[4][3];
    #pragma unroll
    for (int mt = 0; mt < 4; ++mt)
        #pragma unroll
        for (int jj = 0; jj < 3; ++jj) acc[mt][jj] = (v8f){};

    #pragma unroll
    for (int ph = 0; ph < 2; ++ph) {
        __syncthreads();   // previous phase's readers done before overwrite
        // Cooperative, coalesced load of 64 x 128 A-slab into LDS.
        #pragma unroll
        for (int s = 0; s < 4; ++s) {
            const int col = (cseg + s * 8) * 4;
            const float4 vv = *(const float4*)(srcRow + ph * KP_ + col);
            sA[crow][col + 0] = vv.x;
            sA[crow][col + 1] = vv.y;
            sA[crow][col + 2] = vv.z;
            sA[crow][col + 3] = vv.w;
        }
        __syncthreads();

        const float* wbase = Wq + ph * KP_ + khalf;
        #pragma unroll 2
        for (int kk = 0; kk < KP_; kk += 4) {
            // A fragments (16x4 f32 layout): lane L<16 -> row L, K=kk..kk+1;
            // lane L>=16 -> row L-16, K=kk+2..kk+3. One per M-tile.
            v2f a[4];
            #pragma unroll
            for (int mt = 0; mt < 4; ++mt) {
                const float2 av =
                    *(const float2*)(&sA[mt * 16 + lrow][kk + khalf]);
                a[mt].x = av.x; a[mt].y = av.y;
            }
            #pragma unroll
            for (int jj = 0; jj < 3; ++jj) {
                const int n0 = (wave * 3 + jj) * 16;
                // B fragment (4x16): B[kq, n] = Wq[n0+n, k0+kq]; mirrored K split
                const float2 bv =
                    *(const float2*)(wbase + (size_t)(n0 + lrow) * D_ + kk);
                v2f bf; bf.x = bv.x; bf.y = bv.y;
                #pragma unroll
                for (int mt = 0; mt < 4; ++mt)
                    acc[mt][jj] = __builtin_amdgcn_wmma_f32_16x16x4_f32(
                        false, a[mt], false, bf, (short)0, acc[mt][jj],
                        false, false);
            }
        }
    }

    // Store: C layout VGPR i -> lanes 0-15: M=i, N=lane; lanes 16-31: M=i+8.
    const int mhi = (lane >> 4) << 3;
    #pragma unroll
    for (int jj = 0; jj < 3; ++jj) {
        const int n      = (wave * 3 + jj) * 16 + lrow;
        const float bias = bq[n];
        const int which  = n >> 8;           // 0=q 1=k 2=v
        const int hh     = (n & 255) >> 5;   // head
        const int dd     = n & 31;           // within-head dim
        float* base = ws + (size_t)which * QKV_ELEMS;
        #pragma unroll
        for (int mt = 0; mt < 4; ++mt) {
            #pragma unroll
            for (int i = 0; i < 8; ++i) {
                const int m  = m0 + mt * 16 + i + mhi;
                const int bj = m / T_;
                const int t  = m - bj * T_;
                base[(((size_t)bj * H_ + hh) * T_ + t) * DH_ + dd] =
                    acc[mt][jj][i] + bias;
            }
        }
    }
}

// ---------------------------------------------------------------------------
// Kernel 2: windowed attention. One wave per (bj, h, t); lane = d in [0,32).
// Coalesced 128B row loads, shfl_xor tree reduction for the 9 logits,
// register softmax, weighted v-sum. Neighboring waves share k/v windows in
// L0/L2 (k,v arrays ~191MB vs 192MB L2).
// ---------------------------------------------------------------------------
__global__ __launch_bounds__(256)
void local_window_attn(const float* __restrict__ ws,
                       const float* __restrict__ tau,
                       float* __restrict__ out)
{
    const float* qb = ws;
    const float* kb = ws + QKV_ELEMS;
    const float* vb = ws + 2 * QKV_ELEMS;

    const int wg   = blockIdx.x * 8 + (threadIdx.x >> 5);
    const int lane = threadIdx.x & 31;

    const int t   = wg % T_;
    const int rem = wg / T_;
    const int h   = rem % H_;
    const int bj  = rem / H_;

    const size_t rowbase = (size_t)(bj * H_ + h) * T_;

    const float q = qb[(rowbase + t) * DH_ + lane];

    float tc = tau[h];
    tc = tc < 0.001f ? 0.001f : tc;
    const float scale = 1.0f / (5.656854249492381f * tc);   // sqrt(32) * tau

    float logit[KW_], vv[KW_];
    #pragma unroll
    for (int kk = 0; kk < KW_; ++kk) {
        const int  ts    = t + kk - WIN_;
        const bool valid = (ts >= 0) && (ts < T_);
        const int  tsc   = valid ? ts : t;
        const size_t idx = (rowbase + tsc) * DH_ + lane;
        const float kd = kb[idx];
        vv[kk] = vb[idx];
        float s = q * kd;
        #pragma unroll
        for (int off = 16; off >= 1; off >>= 1)
            s += __shfl_xor(s, off, 32);
        logit[kk] = valid ? s * scale : -INFINITY;
    }

    float mx = logit[0];
    #pragma unroll
    for (int kk = 1; kk < KW_; ++kk) mx = fmaxf(mx, logit[kk]);

    float ssum = 0.0f, accv = 0.0f;
    #pragma unroll
    for (int kk = 0; kk < KW_; ++kk) {
        const float p = __expf(logit[kk] - mx);   // exp(-inf) = 0 for OOB
        ssum += p;
        accv += p * vv[kk];
    }
    const float o = accv / ssum;

    const int b = bj / J_;
    const int j = bj - b * J_;
    out[(((size_t)b * T_ + t) * J_ + j) * D_ + h * DH_ + lane] = o;
}

// ---------------------------------------------------------------------------
extern "C" void kernel_launch(void* const* d_in, const int* in_sizes, int n_in,
                              void* d_out, int out_size, void* d_ws, size_t ws_size,
                              hipStream_t stream) {
    const float* x   = (const float*)d_in[0];   // (B, T, J, D) f32
    const float* Wq  = (const float*)d_in[1];   // (3D, D) f32
    const float* bq  = (const float*)d_in[2];   // (3D,) f32
    const float* tau = (const float*)d_in[3];   // (H,) f32
    float* out = (float*)d_out;
    float* ws  = (float*)d_ws;                  // q/k/v staging, 3 * 95.6 MB

    qkv_gemm_wmma<<<M_TOTAL / MB_, 512, 0, stream>>>(x, Wq, bq, ws);
    local_window_attn<<<(BJ_ * H_ * T_) / 8, 256, 0, stream>>>(ws, tau, out);
}